// CellListNL_39539468927503
// MI455X (gfx1250) — compile-verified
//
#include <hip/hip_runtime.h>
#include <hip/hip_bf16.h>

#define RC   5.0f
#define RC2  25.0f
#define BSZ  128          // structures
#define MSZ  256          // atoms per structure
#define NATOMS (BSZ * MSZ)
#define NROWS  (BSZ * MSZ)
#define PCAP 2097152      // fixed pair capacity

typedef float v2f __attribute__((ext_vector_type(2)));
typedef float v8f __attribute__((ext_vector_type(8)));

// broadcast from lane l0 (if half==0) or lane l1 (if half==1); l0/l1 compile-time constants
__device__ __forceinline__ int bcast2i(int x, int l0, int l1, int half) {
    int lo = __builtin_amdgcn_readlane(x, l0);
    int hi = __builtin_amdgcn_readlane(x, l1);
    return half ? hi : lo;
}
__device__ __forceinline__ float bcast2f(float x, int l0, int l1, int half) {
    return __int_as_float(bcast2i(__float_as_int(x), l0, l1, half));
}

// ---------------- init: min accumulator ----------------
__global__ void k_init_min(int* wsmin) {
    if (threadIdx.x < 3) wsmin[threadIdx.x] = 0x7F7FFFFF;  // +FLT_MAX bits
}

// ---------------- global per-axis min (coords >= 0 so float bits are order-preserving ints)
__global__ void k_min(const float* __restrict__ coord, int* __restrict__ wsmin) {
    int n = blockIdx.x * blockDim.x + threadIdx.x;
    if (n < NATOMS) {
        atomicMin(&wsmin[0], __float_as_int(coord[3 * n + 0]));
        atomicMin(&wsmin[1], __float_as_int(coord[3 * n + 1]));
        atomicMin(&wsmin[2], __float_as_int(coord[3 * n + 2]));
    }
}

// ---------------- cell binning ----------------
__global__ void k_cell(const float* __restrict__ coord, const int* __restrict__ wsmin,
                       int* __restrict__ cells) {
    int n = blockIdx.x * blockDim.x + threadIdx.x;
    if (n < NATOMS) {
        float mx = __int_as_float(wsmin[0]);
        float my = __int_as_float(wsmin[1]);
        float mz = __int_as_float(wsmin[2]);
        cells[3 * n + 0] = (int)floorf((coord[3 * n + 0] - mx) * (1.0f / RC));
        cells[3 * n + 1] = (int)floorf((coord[3 * n + 1] - my) * (1.0f / RC));
        cells[3 * n + 2] = (int)floorf((coord[3 * n + 2] - mz) * (1.0f / RC));
    }
}

// ---------------- zero-fill output (padding entries are all zeros) ----------------
__global__ void k_zero(uint4* __restrict__ p, int n4) {
    int i = blockIdx.x * blockDim.x + threadIdx.x;
    if (i < n4) p[i] = make_uint4(0u, 0u, 0u, 0u);
}

// ---------------- pair kernel: one wave per (structure b, i-tile of 16 rows) ----------------
// WMMA computes sq_ij = |p_i|^2 + |p_j|^2 - 2 p_i . p_j for a 16x16 tile:
//   A (16x4) rows: (-2x_i, -2y_i, -2z_i, 1) ; B (4x16) cols: (x_j, y_j, z_j, |p_j|^2) ; C = |p_i|^2
template <bool WRITE>
__global__ __launch_bounds__(256) void k_pairs(
        const float* __restrict__ coord, const int* __restrict__ cells,
        int* __restrict__ counts, const int* __restrict__ offsets,
        int* __restrict__ out_ind, float* __restrict__ out_d, float* __restrict__ out_df) {
    const int w     = threadIdx.x >> 5;
    const int lane  = threadIdx.x & 31;
    const int half  = lane >> 4;
    const int m     = lane & 15;
    const int wtask = blockIdx.x * 8 + w;          // 2048 wave tasks
    const int b     = wtask >> 4;
    const int i0    = (wtask & 15) * 16;           // first i-row of this tile
    const int rowBase = b * MSZ + i0;              // flat (b,i) row index == global atom index

    // ---- owner lanes 0..15 hold i-row data for row = lane ----
    float xi = 0.f, yi = 0.f, zi = 0.f, r2i = 0.f;
    int cix = 0, ciy = 0, ciz = 0;
    int runOff = 0;                                 // lane r (<16) owns running offset of row r
    if (lane < 16) {
        int gi = rowBase + lane;
        xi = coord[3 * gi + 0];
        yi = coord[3 * gi + 1];
        zi = coord[3 * gi + 2];
        r2i = xi * xi + yi * yi + zi * zi;
        cix = cells[3 * gi + 0];
        ciy = cells[3 * gi + 1];
        ciz = cells[3 * gi + 2];
        if (WRITE) runOff = offsets[gi];
    }

    // ---- A matrix (16x4 f32): lanes0-15 hold K0,K1; lanes16-31 hold K2,K3 for row m ----
    v2f a;
    a.x = half ? (-2.0f * __shfl(zi, m, 32)) : (-2.0f * __shfl(xi, m, 32));
    a.y = half ? 1.0f : (-2.0f * __shfl(yi, m, 32));

    // ---- C matrix (16x16 f32 acc): VGPR v -> rows v (lanes0-15) and v+8 (lanes16-31) ----
    v8f c;
#pragma unroll
    for (int v = 0; v < 8; ++v) c[v] = bcast2f(r2i, v, v + 8, half);

    // ---- hoisted per-v row data (invariant across j-tiles): my row r = v + 8*half ----
    float rxi[8], ryi[8], rzi[8];
    int rcx[8], rcy[8], rcz[8];
#pragma unroll
    for (int v = 0; v < 8; ++v) {
        rxi[v] = bcast2f(xi, v, v + 8, half);
        ryi[v] = bcast2f(yi, v, v + 8, half);
        rzi[v] = bcast2f(zi, v, v + 8, half);
        rcx[v] = bcast2i(cix, v, v + 8, half);
        rcy[v] = bcast2i(ciy, v, v + 8, half);
        rcz[v] = bcast2i(ciz, v, v + 8, half);
    }

    for (int jt = 0; jt < 16; ++jt) {
        const int gj = b * MSZ + jt * 16 + m;      // this lane's column atom
        const float xj = coord[3 * gj + 0];
        const float yj = coord[3 * gj + 1];
        const float zj = coord[3 * gj + 2];
        const float r2j = xj * xj + yj * yj + zj * zj;
        const int cjx = cells[3 * gj + 0];
        const int cjy = cells[3 * gj + 1];
        const int cjz = cells[3 * gj + 2];

        if (jt < 15) {  // prefetch next j-tile (uniform branch) -> global_prefetch_b8
            __builtin_prefetch(&coord[3 * (gj + 16)], 0, 3);
            __builtin_prefetch(&cells[3 * (gj + 16)], 0, 3);
        }

        // ---- B matrix (4x16 f32): lanes0-15 hold K0,K1; lanes16-31 hold K2,K3 for col m ----
        v2f bb;
        bb.x = half ? zj : xj;
        bb.y = half ? r2j : yj;

        v8f d = __builtin_amdgcn_wmma_f32_16x16x4_f32(
            /*neg_a=*/false, a, /*neg_b=*/false, bb,
            /*c_mod=*/(short)0, c, /*reuse_a=*/false, /*reuse_b=*/false);

#pragma unroll
        for (int v = 0; v < 8; ++v) {
            const int r = v + 8 * half;            // my row within the i-tile for this VGPR
            const int offr = bcast2i(runOff, v, v + 8, half);

            const float sq = d[v];
            const int gi = rowBase + r;
            const int dx = rcx[v] - cjx, dy = rcy[v] - cjy, dz = rcz[v] - cjz;
            const bool adj = (dx <= 1 && dx >= -1) && (dy <= 1 && dy >= -1) && (dz <= 1 && dz >= -1);
            const bool ok = adj && (gi != gj) && (sq > 0.0f) && (sq < RC2);

            const unsigned bal = (unsigned)__ballot(ok);
            const unsigned below = bal & ((1u << lane) - 1u);
            const unsigned hmask = half ? 0xFFFF0000u : 0x0000FFFFu;
            const int p = offr + __popc(below & hmask);

            if (WRITE && ok && p < PCAP) {
                out_ind[2 * p + 0] = gi;
                out_ind[2 * p + 1] = gj;
                out_d[p] = sqrtf(sq);
                out_df[3 * p + 0] = xj - rxi[v];
                out_df[3 * p + 1] = yj - ryi[v];
                out_df[3 * p + 2] = zj - rzi[v];
            }
            // owner lanes advance their row's running offset (j-order preserved)
            const int cntLo = __popc(bal & 0xFFFFu);
            const int cntHi = __popc(bal >> 16);
            if (lane == v)     runOff += cntLo;
            if (lane == v + 8) runOff += cntHi;
        }
    }

    if (!WRITE && lane < 16) counts[rowBase + lane] = runOff;
}

// ---------------- single-block exclusive scan over 32768 row counts ----------------
__global__ __launch_bounds__(1024) void k_scan(const int* __restrict__ counts,
                                               int* __restrict__ offsets) {
    __shared__ int sm[1024];
    __shared__ int s_total;
    const int tid = threadIdx.x;
    if (tid == 0) s_total = 0;
    __syncthreads();
    for (int chunk = 0; chunk < NROWS; chunk += 1024) {
        const int v = counts[chunk + tid];
        sm[tid] = v;
        __syncthreads();
#pragma unroll
        for (int dstep = 1; dstep < 1024; dstep <<= 1) {
            int t = (tid >= dstep) ? sm[tid - dstep] : 0;
            __syncthreads();
            sm[tid] += t;
            __syncthreads();
        }
        const int incl = sm[tid];
        const int base = s_total;
        offsets[chunk + tid] = base + incl - v;    // exclusive prefix
        __syncthreads();
        if (tid == 1023) s_total = base + incl;
        __syncthreads();
    }
}

extern "C" void kernel_launch(void* const* d_in, const int* in_sizes, int n_in,
                              void* d_out, int out_size, void* d_ws, size_t ws_size,
                              hipStream_t stream) {
    const float* coord = (const float*)d_in[0];
    // d_in[1] (ind_1) is structural (contiguous blocks of M) — not needed.

    // workspace layout (ints)
    int* wsi     = (int*)d_ws;
    int* wsmin   = wsi;                  // 3 (+1 pad)
    int* cells   = wsi + 4;              // NATOMS*3
    int* counts  = cells + NATOMS * 3;   // NROWS
    int* offsets = counts + NROWS;       // NROWS

    // output layout: ind_2 [P,2] i32 | d [P] f32 | df [P,3] f32
    int*   out_ind = (int*)d_out;
    float* out_d   = (float*)d_out + 2 * PCAP;
    float* out_df  = (float*)d_out + 3 * PCAP;

    k_init_min<<<1, 64, 0, stream>>>(wsmin);
    k_min<<<(NATOMS + 255) / 256, 256, 0, stream>>>(coord, wsmin);
    k_cell<<<(NATOMS + 255) / 256, 256, 0, stream>>>(coord, wsmin, cells);

    const int n4 = (6 * PCAP) / 4;       // total output dwords / 4
    k_zero<<<(n4 + 255) / 256, 256, 0, stream>>>((uint4*)d_out, n4);

    k_pairs<false><<<256, 256, 0, stream>>>(coord, cells, counts, nullptr,
                                            nullptr, nullptr, nullptr);
    k_scan<<<1, 1024, 0, stream>>>(counts, offsets);
    k_pairs<true><<<256, 256, 0, stream>>>(coord, cells, nullptr, offsets,
                                           out_ind, out_d, out_df);
}